// KineticPooling_5257039970557
// MI455X (gfx1250) — compile-verified
//
#include <hip/hip_runtime.h>

// Problem constants (from reference)
#define NUP    16
#define NMO    64
#define NCONF  64
#define NBATCH 4096
#define NELEC  32   // NUP + NDOWN

typedef __attribute__((ext_vector_type(2))) float v2f;
typedef __attribute__((ext_vector_type(4))) float v4f;
typedef __attribute__((ext_vector_type(8))) float v8f;

// ---------------------------------------------------------------------------
// Kernel 1: Bsum = d2MO + 2*dJdMO + d2JMO   (8.4M floats, float4-vectorized)
// ---------------------------------------------------------------------------
__global__ __launch_bounds__(256) void bsum_kernel(
    const float* __restrict__ d2MO, const float* __restrict__ dJdMO,
    const float* __restrict__ d2JMO, float* __restrict__ Bsum, int n4) {
  int i = blockIdx.x * blockDim.x + threadIdx.x;
  if (i < n4) {
    v4f a = ((const v4f*)d2MO)[i];
    v4f b = ((const v4f*)dJdMO)[i];
    v4f c = ((const v4f*)d2JMO)[i];
    ((v4f*)Bsum)[i] = a + 2.0f * b + c;
  }
}

// ---------------------------------------------------------------------------
// Kernel 2: one wave32 per (config, batch) pair.
//   lanes 0-15  : rows 0-15 of the spin-up   16x16 matrices
//   lanes 16-31 : rows 0-15 of the spin-down 16x16 matrices
// Gauss-Jordan inverse (row-per-lane, shfl pivot broadcast), det = prod pivots,
// then tr(InvUp*Bup)+tr(InvDn*Bdn) via 8 chained V_WMMA_F32_16X16X4_F32.
// ---------------------------------------------------------------------------
__global__ __launch_bounds__(256) void kinetic_kernel(
    const float* __restrict__ MO, const float* __restrict__ Bsum,
    const int* __restrict__ cup, const int* __restrict__ cdown,
    float* __restrict__ out) {
  const int lane = threadIdx.x & 31;
  const int half = lane >> 4;     // 0 = up, 1 = down
  const int r    = lane & 15;     // row within spin matrix
  const int wid  = (blockIdx.x * blockDim.x + threadIdx.x) >> 5;
  const int c = wid & (NCONF - 1);        // config fastest -> MO rows of batch b
  const int b = wid >> 6;                 // stay hot in L0/L2 across 64 waves

  // Column indices: lanes 0-15 hold cup[c][lane], lanes 16-31 cdown[c][lane-16]
  int iv = half ? cdown[c * 16 + r] : cup[c * 16 + r];

  // Row-per-lane gather of A (global row index == lane for this layout)
  const float* mrow = MO + (size_t)b * (NELEC * NMO) + lane * NMO;
  float a[16], inv[16];
#pragma unroll
  for (int j = 0; j < 16; ++j) {
    int cj = __shfl(iv, (lane & 16) | j, 32);   // own spin-group's j-th column
    a[j]   = mrow[cj];
    inv[j] = (j == r) ? 1.0f : 0.0f;
  }

  // ---- Gauss-Jordan inversion (no pivoting; matrices generically nonsingular)
  // After step k: A columns <k are e-vectors, Inv columns >k are still identity,
  // so A updates need only j>=k and Inv updates only j<k (+ column k closed form).
  float det = 1.0f;
#pragma unroll
  for (int k = 0; k < 16; ++k) {
    const int src = (lane & 16) | k;            // pivot-row lane of own spin
    float pivot = __shfl(a[k], src, 32);
    det *= pivot;                               // lanes<16 -> det_up, >=16 -> det_dn
    float pinv = 1.0f / pivot;
    float m = a[k];                             // eliminate multiplier (pre-update)
    bool isPiv = (r == k);
#pragma unroll
    for (int j = k; j < 16; ++j) {              // A: only columns >= k change
      float rk = __shfl(a[j], src, 32) * pinv;
      a[j] = isPiv ? rk : fmaf(-m, rk, a[j]);
    }
#pragma unroll
    for (int j = 0; j < k; ++j) {               // Inv: only columns < k are nontrivial
      float ri = __shfl(inv[j], src, 32) * pinv;
      inv[j] = isPiv ? ri : fmaf(-m, ri, inv[j]);
    }
    // Inv column k: previous Inv[:,k] = delta, Inv[k][k] = 1 -> closed form
    inv[k] = isPiv ? pinv : -m * pinv;
  }

  // ---- Trace via WMMA: acc = [InvUp | InvDn] (16x32) x [Bup ; Bdn] (32x16)
  // Diagonal of acc = per-row contributions to tr_up + tr_down.
  const int s  = lane & 15;
  int cu = __shfl(iv, s, 32);                   // up-column index for N = s
  int cd = __shfl(iv, 16 | s, 32);              // down-column index for N = s
  const float* brow = Bsum + (size_t)b * (NELEC * NMO);

  v8f acc = {};
#pragma unroll
  for (int t = 0; t < 4; ++t) {                 // spin-up, K-chunk 4t..4t+3
    // A layout (f32 16x4): lanes 0-15 K=4t+{0,1}; lanes 16-31 K=4t+{2,3}; M = s
    float lo0 = __shfl(inv[4 * t + 0], s, 32);
    float hi0 = __shfl(inv[4 * t + 2], s, 32);
    float lo1 = __shfl(inv[4 * t + 1], s, 32);
    float hi1 = __shfl(inv[4 * t + 3], s, 32);
    v2f A; A.x = half ? hi0 : lo0; A.y = half ? hi1 : lo1;
    // B layout (f32 4x16): VGPR0 rows 4t+{0,1}, VGPR1 rows 4t+{2,3}; N = s
    v2f B;
    B.x = brow[(4 * t + half) * NMO + cu];
    B.y = brow[(4 * t + 2 + half) * NMO + cu];
    acc = __builtin_amdgcn_wmma_f32_16x16x4_f32(false, A, false, B,
                                                (short)0, acc, false, false);
  }
#pragma unroll
  for (int t = 0; t < 4; ++t) {                 // spin-down, K-chunk 4t..4t+3
    float lo0 = __shfl(inv[4 * t + 0], 16 | s, 32);
    float hi0 = __shfl(inv[4 * t + 2], 16 | s, 32);
    float lo1 = __shfl(inv[4 * t + 1], 16 | s, 32);
    float hi1 = __shfl(inv[4 * t + 3], 16 | s, 32);
    v2f A; A.x = half ? hi0 : lo0; A.y = half ? hi1 : lo1;
    v2f B;
    B.x = brow[(16 + 4 * t + half) * NMO + cd];
    B.y = brow[(16 + 4 * t + 2 + half) * NMO + cd];
    acc = __builtin_amdgcn_wmma_f32_16x16x4_f32(false, A, false, B,
                                                (short)0, acc, false, false);
  }

  // ---- Diagonal extraction: C/D layout -> D[rr][rr] at lane rr (M=rr) and
  // D[8+rr][8+rr] at lane 24+rr, both in VGPR rr.
  float tsum = 0.0f;
#pragma unroll
  for (int rr = 0; rr < 8; ++rr) {
    bool dsel = (lane == rr) || (lane == 24 + rr);
    tsum += dsel ? acc[rr] : 0.0f;
  }
#pragma unroll
  for (int off = 16; off > 0; off >>= 1)
    tsum += __shfl_xor(tsum, off, 32);          // tr_up + tr_down in all lanes

  float det_u = __shfl(det, 0, 32);
  float det_d = __shfl(det, 16, 32);
  float dp = det_u * det_d;
  if (lane == 0) {
    size_t o = (size_t)b * NCONF + c;
    out[o] = -0.5f * tsum * dp;                         // kinetic (B, NCONF)
    out[(size_t)NBATCH * NCONF + o] = dp;               // det_prod (B, NCONF)
  }
}

// ---------------------------------------------------------------------------
extern "C" void kernel_launch(void* const* d_in, const int* in_sizes, int n_in,
                              void* d_out, int out_size, void* d_ws, size_t ws_size,
                              hipStream_t stream) {
  const float* MO    = (const float*)d_in[0];
  const float* d2MO  = (const float*)d_in[1];
  const float* dJdMO = (const float*)d_in[2];
  const float* d2JMO = (const float*)d_in[3];
  const int*   cup   = (const int*)d_in[4];
  const int*   cdn   = (const int*)d_in[5];
  float* out  = (float*)d_out;
  float* Bsum = (float*)d_ws;   // 32 MB scratch

  const int n4 = (NBATCH * NELEC * NMO) / 4;        // 2,097,152 float4s
  bsum_kernel<<<n4 / 256, 256, 0, stream>>>(d2MO, dJdMO, d2JMO, Bsum, n4);

  const int waves = NBATCH * NCONF;                 // 262,144 waves
  kinetic_kernel<<<(waves * 32) / 256, 256, 0, stream>>>(MO, Bsum, cup, cdn, out);
}